// H2GCN_87205015978221
// MI455X (gfx1250) — compile-verified
//
#include <hip/hip_runtime.h>
#include <cmath>

#define N_ENT   100000
#define N_REL   50
#define N_EDGE  200000
#define NNZ_    1000000
#define DD      128
#define BATCH_  4096

typedef float v2f __attribute__((ext_vector_type(2)));
typedef float v8f __attribute__((ext_vector_type(8)));

// Hardware f32 atomic add (global_atomic_add_f32), avoiding any CAS expansion.
__device__ __forceinline__ void fatomic_add(float* p, float v) {
  unsafeAtomicAdd(p, v);
}

// ---------------------------------------------------------------------------
// Kernel 1: Y = emb_E @ W^T + b   (raw pre-lorentz result)
// WMMA f32 16x16x4. Each wave computes a 16-row x 128-col strip.
// W is staged transposed in LDS: Wt[k*128+j] = W[j*128+k], so the B fragment
// (B[k][n] = W[n][k]) reads consecutive addresses across lanes (no conflicts).
// ---------------------------------------------------------------------------
__global__ __launch_bounds__(256) void gemm_kernel(
    const float* __restrict__ X, const float* __restrict__ W,
    const float* __restrict__ bias, float* __restrict__ Y) {
  __shared__ float Wt[DD * DD];  // 64 KB
  const int tid = threadIdx.x;
  for (int i = tid; i < DD * DD; i += 256) {
    int j = i >> 7, k = i & 127;
    Wt[k * DD + j] = W[i];
  }
  __syncthreads();

  const int wave = tid >> 5, lane = tid & 31;
  const int tileRow = blockIdx.x * 128 + wave * 16;
  const int m    = lane & 15;          // row (A) / col (B) within tile
  const int koff = (lane >> 4) << 1;   // lanes 16..31 hold K+2,K+3
  const int rowA = tileRow + m;
  const int rA   = rowA < N_ENT ? rowA : N_ENT - 1;
  const float* Arow = X + (long)rA * DD;

  v8f acc[8] = {};
  for (int k0 = 0; k0 < DD; k0 += 4) {
    const int kb = k0 + koff;
    v2f a;
    a.x = Arow[kb];
    a.y = Arow[kb + 1];
#pragma unroll
    for (int t = 0; t < 8; ++t) {
      const int n = t * 16 + m;
      v2f b;
      b.x = Wt[kb * DD + n];
      b.y = Wt[(kb + 1) * DD + n];
      acc[t] = __builtin_amdgcn_wmma_f32_16x16x4_f32(
          false, a, false, b, (short)0, acc[t], false, false);
    }
  }

  // C layout: VGPR r, lanes 0-15 -> row r, lanes 16-31 -> row r+8; col = lane&15.
  const int rbase = tileRow + ((lane >> 4) << 3);
  if (tileRow + 16 <= N_ENT) {
    // Fast path: whole tile in range, unguarded coalesced stores.
#pragma unroll
    for (int t = 0; t < 8; ++t) {
      const int col = t * 16 + m;
      const float bv = bias[col];
      float* dst = Y + (long)rbase * DD + col;
#pragma unroll
      for (int r = 0; r < 8; ++r) {
        dst[(long)r * DD] = acc[t][r] + bv;
      }
    }
  } else {
#pragma unroll
    for (int t = 0; t < 8; ++t) {
      const int col = t * 16 + m;
      const float bv = bias[col];
#pragma unroll
      for (int r = 0; r < 8; ++r) {
        const int row = rbase + r;
        if (row < N_ENT) Y[(long)row * DD + col] = acc[t][r] + bv;
      }
    }
  }
}

// ---------------------------------------------------------------------------
// Kernel 2: lorentz_linear nonlinearity, in place. One wave per row.
// ---------------------------------------------------------------------------
__global__ __launch_bounds__(256) void lorentz_kernel(
    float* __restrict__ Y, const float* __restrict__ scale_p) {
  const int wave = threadIdx.x >> 5, lane = threadIdx.x & 31;
  const int row = blockIdx.x * 8 + wave;
  if (row >= N_ENT) return;
  float* yr = Y + (long)row * DD;

  float v[4];
  float local = 0.f;
#pragma unroll
  for (int k = 0; k < 4; ++k) {
    const int c = lane + k * 32;
    v[k] = yr[c];
    if (c != 0) local += v[k] * v[k];
  }
#pragma unroll
  for (int off = 16; off; off >>= 1) local += __shfl_xor(local, off, 32);
  const float y0 = __shfl(v[0], 0, 32);

  const float es   = expf(scale_p[0]);
  const float time = es / (1.f + expf(-y0)) + 1.1f;
  const float sqn  = fmaxf(local, 1e-8f);
  const float s    = sqrtf((time * time - 1.f) / sqn);
#pragma unroll
  for (int k = 0; k < 4; ++k) {
    const int c = lane + k * 32;
    yr[c] = (c == 0) ? time : v[k] * s;
  }
}

// ---------------------------------------------------------------------------
// Kernel 3: Xe[edges[i]] += Xl[vertex[i]] - emb_ty[ty[i]].  Wave per nonzero.
// ---------------------------------------------------------------------------
__global__ __launch_bounds__(256) void scatter1_kernel(
    const float* __restrict__ Xl, const float* __restrict__ emb_ty,
    const int* __restrict__ vertex, const int* __restrict__ edges,
    const int* __restrict__ ty, float* __restrict__ Xe) {
  const int wave = threadIdx.x >> 5, lane = threadIdx.x & 31;
  const long idx = (long)blockIdx.x * 8 + wave;
  if (idx >= NNZ_) return;
  const int v = vertex[idx], e = edges[idx], t = ty[idx];
  const float4 a = ((const float4*)(Xl + (long)v * DD))[lane];
  const float4 b = ((const float4*)(emb_ty + (long)t * DD))[lane];
  float* dst = Xe + (long)e * DD + lane * 4;
  fatomic_add(dst + 0, a.x - b.x);
  fatomic_add(dst + 1, a.y - b.y);
  fatomic_add(dst + 2, a.z - b.z);
  fatomic_add(dst + 3, a.w - b.w);
}

// ---------------------------------------------------------------------------
// Kernel 4: Xv[vertex[i]] += Xe[edges[i]].  Wave per nonzero.
// ---------------------------------------------------------------------------
__global__ __launch_bounds__(256) void scatter2_kernel(
    const float* __restrict__ Xe, const int* __restrict__ vertex,
    const int* __restrict__ edges, float* __restrict__ Xv) {
  const int wave = threadIdx.x >> 5, lane = threadIdx.x & 31;
  const long idx = (long)blockIdx.x * 8 + wave;
  if (idx >= NNZ_) return;
  const int v = vertex[idx], e = edges[idx];
  const float4 a = ((const float4*)(Xe + (long)e * DD))[lane];
  float* dst = Xv + (long)v * DD + lane * 4;
  fatomic_add(dst + 0, a.x);
  fatomic_add(dst + 1, a.y);
  fatomic_add(dst + 2, a.z);
  fatomic_add(dst + 3, a.w);
}

// ---------------------------------------------------------------------------
// Kernel 5: E_e = logmap0(eps*Xv + Xl, k=1), col0=0, row0=1.0.
// In place over the Xv buffer. One wave per row.
// ---------------------------------------------------------------------------
__global__ __launch_bounds__(256) void finalize_kernel(
    const float* __restrict__ Xl, float* __restrict__ Xv,
    const float* __restrict__ eps_p) {
  const int wave = threadIdx.x >> 5, lane = threadIdx.x & 31;
  const int row = blockIdx.x * 8 + wave;
  if (row >= N_ENT) return;
  const float eps = eps_p[0];
  const float* xl = Xl + (long)row * DD;
  float* xv = Xv + (long)row * DD;

  float u[4];
  float local = 0.f;
#pragma unroll
  for (int k = 0; k < 4; ++k) {
    const int c = lane + k * 32;
    u[k] = eps * xv[c] + xl[c];
    if (c != 0) local += u[k] * u[k];
  }
#pragma unroll
  for (int off = 16; off; off >>= 1) local += __shfl_xor(local, off, 32);
  const float u0 = __shfl(u[0], 0, 32);

  const float theta = fmaxf(u0, 1.0f + 1e-7f);
  const float ynorm = fmaxf(sqrtf(local), 1e-8f);
  const float fac   = acoshf(theta) / ynorm;
#pragma unroll
  for (int k = 0; k < 4; ++k) {
    const int c = lane + k * 32;
    float val = (c == 0) ? 0.f : u[k] * fac;
    if (row == 0) val = 1.f;
    xv[c] = val;
  }
}

// ---------------------------------------------------------------------------
// Kernel 6: out[b] = sum_c E[e1]*E[e2]*E[e3]*E[e4]*E[e5]*E[e6]*R[r].
// Row 0 of R is treated as all-ones. One wave per batch element.
// ---------------------------------------------------------------------------
__global__ __launch_bounds__(256) void score_kernel(
    const float* __restrict__ E, const float* __restrict__ R,
    const int* __restrict__ r_idx,
    const int* __restrict__ i1, const int* __restrict__ i2,
    const int* __restrict__ i3, const int* __restrict__ i4,
    const int* __restrict__ i5, const int* __restrict__ i6,
    float* __restrict__ out) {
  const int wave = threadIdx.x >> 5, lane = threadIdx.x & 31;
  const int b = blockIdx.x * 8 + wave;
  if (b >= BATCH_) return;

  const float4* E4 = (const float4*)E;
  float4 p = E4[(long)i1[b] * 32 + lane];
  const int idxs[5] = {i2[b], i3[b], i4[b], i5[b], i6[b]};
#pragma unroll
  for (int q = 0; q < 5; ++q) {
    const float4 e = E4[(long)idxs[q] * 32 + lane];
    p.x *= e.x; p.y *= e.y; p.z *= e.z; p.w *= e.w;
  }
  const int r = r_idx[b];
  if (r != 0) {
    const float4 rv = ((const float4*)R)[(long)r * 32 + lane];
    p.x *= rv.x; p.y *= rv.y; p.z *= rv.z; p.w *= rv.w;
  }
  float sum = p.x + p.y + p.z + p.w;
#pragma unroll
  for (int off = 16; off; off >>= 1) sum += __shfl_xor(sum, off, 32);
  if (lane == 0) out[b] = sum;
}

// ---------------------------------------------------------------------------
extern "C" void kernel_launch(void* const* d_in, const int* in_sizes, int n_in,
                              void* d_out, int out_size, void* d_ws, size_t ws_size,
                              hipStream_t stream) {
  const float* emb_E     = (const float*)d_in[0];
  const float* emb_R     = (const float*)d_in[1];
  const float* emb_ty    = (const float*)d_in[2];
  const float* lin_W     = (const float*)d_in[3];
  const float* lin_b     = (const float*)d_in[4];
  const float* lin_scale = (const float*)d_in[5];
  const float* eps       = (const float*)d_in[6];
  // d_in[7] = ms (unused by reference output)
  const int* r_idx = (const int*)d_in[8];
  const int* e1    = (const int*)d_in[9];
  const int* e2    = (const int*)d_in[10];
  const int* e3    = (const int*)d_in[11];
  const int* e4    = (const int*)d_in[12];
  const int* e5    = (const int*)d_in[13];
  const int* e6    = (const int*)d_in[14];
  const int* vertex = (const int*)d_in[15];
  const int* edges  = (const int*)d_in[16];
  const int* ty     = (const int*)d_in[17];
  float* out = (float*)d_out;

  // Workspace layout (floats): Xl[N_ENT*DD] | Xe[N_EDGE*DD] | Xv[N_ENT*DD]
  float* Xl = (float*)d_ws;
  float* Xe = Xl + (size_t)N_ENT * DD;
  float* Xv = Xe + (size_t)N_EDGE * DD;

  // Zero the segment-sum accumulators (Xe and Xv are contiguous).
  hipMemsetAsync(Xe, 0, (size_t)(N_EDGE + N_ENT) * DD * sizeof(float), stream);

  gemm_kernel<<<(N_ENT + 127) / 128, 256, 0, stream>>>(emb_E, lin_W, lin_b, Xl);
  lorentz_kernel<<<(N_ENT + 7) / 8, 256, 0, stream>>>(Xl, lin_scale);
  scatter1_kernel<<<(NNZ_ + 7) / 8, 256, 0, stream>>>(Xl, emb_ty, vertex, edges, ty, Xe);
  scatter2_kernel<<<(NNZ_ + 7) / 8, 256, 0, stream>>>(Xe, vertex, edges, Xv);
  finalize_kernel<<<(N_ENT + 7) / 8, 256, 0, stream>>>(Xl, Xv, eps);
  score_kernel<<<(BATCH_ + 7) / 8, 256, 0, stream>>>(Xv, emb_R, r_idx,
                                                     e1, e2, e3, e4, e5, e6, out);
}